// coarse_tangent_loss_80418967650822
// MI455X (gfx1250) — compile-verified
//
#include <hip/hip_runtime.h>

typedef __attribute__((ext_vector_type(2))) float v2f;
typedef __attribute__((ext_vector_type(8))) float v8f;

#define KS        7          // conv kernel size
#define NTAPS     49         // 7*7
#define KSTEPS    13         // 13*4 = 52 >= 49 (zero-padded K)
#define NF        14         // directional filters
#define NPAD      16         // channel dim padded for WMMA N
#define PTILE     8          // pooled tile is 8x8 per block
#define CTILE     17         // conv tile 17x17 (covers 8 pooled px, stride 2, win 3)
#define CPIX      (CTILE*CTILE)   // 289
#define CPIX_PAD  304        // 19 * 16
#define NJOBS     19         // M-tiles of 16 pixels per matrix
#define ITILE     23         // input tile (17 + 6 halo)
#define ISTRIDE   24
#define NWAVES    4
#define BLOCK     128
#define IMGW      512
#define TILES_1D  32         // 256 pooled / 8
#define NBLOCKS   (16*TILES_1D*TILES_1D)   // 16384

__global__ __launch_bounds__(BLOCK) void ctl_main(const float* __restrict__ pred,
                                                  const float* __restrict__ gt,
                                                  const float* __restrict__ filt,
                                                  float* __restrict__ partials) {
  __shared__ float sInP[ITILE * ISTRIDE];
  __shared__ float sInG[ITILE * ISTRIDE];
  __shared__ float sConvP[CPIX_PAD * NPAD];
  __shared__ float sConvG[CPIX_PAD * NPAD];
  __shared__ float sRed[BLOCK];

  const int b   = blockIdx.x;
  const int img = b >> 10;           // 1024 tiles per image
  const int rem = b & 1023;
  const int tY  = rem >> 5, tX = rem & 31;
  const int pr0 = tY * PTILE, pc0 = tX * PTILE;        // pooled tile origin
  const int cr0 = 2 * pr0 - 1, cc0 = 2 * pc0 - 1;      // conv tile origin
  const int ir0 = cr0 - 3,  ic0 = cc0 - 3;             // input tile origin
  const int tid = threadIdx.x;
  const size_t ibase = (size_t)img * IMGW * IMGW;

  // ---- stage input tiles (zero pad = conv 'same' padding) ----
  for (int i = tid; i < ITILE * ITILE; i += BLOCK) {
    int ry = i / ITILE, rx = i - ry * ITILE;
    int gy = ir0 + ry, gx = ic0 + rx;
    bool ok = (gy >= 0) & (gy < IMGW) & (gx >= 0) & (gx < IMGW);
    size_t off = ibase + (size_t)gy * IMGW + gx;
    sInP[ry * ISTRIDE + rx] = ok ? pred[off] : 0.0f;
    sInG[ry * ISTRIDE + rx] = ok ? gt[off]   : 0.0f;
  }

  // ---- per-lane B fragments: filters [52 x 16], resident in registers ----
  const int lane = tid & 31;
  const int n    = lane & 15;          // output channel (N)
  const int koff = (lane >> 4) * 2;    // lanes 16-31 hold K+2,K+3
  v2f bfrag[KSTEPS];
#pragma unroll
  for (int kb = 0; kb < KSTEPS; ++kb) {
    int k0 = kb * 4 + koff;
    bfrag[kb][0] = (n < NF && k0     < NTAPS) ? filt[n * NTAPS + k0]     : 0.0f;
    bfrag[kb][1] = (n < NF && k0 + 1 < NTAPS) ? filt[n * NTAPS + k0 + 1] : 0.0f;
  }

  __syncthreads();

  // ---- implicit-GEMM conv: 38 M-tile jobs (19 pred + 19 gt) over 4 waves ----
  const int wave = tid >> 5;
  const int half = lane >> 4;
  for (int j = wave; j < 2 * NJOBS; j += NWAVES) {
    const bool  isG   = (j >= NJOBS);
    const int   jj    = isG ? (j - NJOBS) : j;
    const float* sIn  = isG ? sInG : sInP;
    float*      sConv = isG ? sConvG : sConvP;

    int m = jj * 16 + (lane & 15);               // A-row pixel for this lane
    if (m >= CPIX) m = CPIX - 1;                 // clamp tail (rows unused)
    const int py = m / CTILE, px = m - (m / CTILE) * CTILE;

    v8f acc = {};
#pragma unroll
    for (int kb = 0; kb < KSTEPS; ++kb) {
      int k0 = kb * 4 + koff;
      float a0 = 0.0f, a1 = 0.0f;
      if (k0 < NTAPS)     { int ky = k0 / KS, kx = k0 - ky * KS;
                            a0 = sIn[(py + ky) * ISTRIDE + (px + kx)]; }
      if (k0 + 1 < NTAPS) { int k1 = k0 + 1; int ky = k1 / KS, kx = k1 - ky * KS;
                            a1 = sIn[(py + ky) * ISTRIDE + (px + kx)]; }
      v2f a; a[0] = a0; a[1] = a1;
      acc = __builtin_amdgcn_wmma_f32_16x16x4_f32(
          /*neg_a=*/false, a, /*neg_b=*/false, bfrag[kb],
          /*c_mod=*/(short)0, acc, /*reuse_a=*/false, /*reuse_b=*/false);
    }
    // D layout: VGPR r holds row r (lanes 0-15) / row r+8 (lanes 16-31)
#pragma unroll
    for (int r = 0; r < 8; ++r) {
      int mrow = jj * 16 + r + 8 * half;
      sConv[mrow * NPAD + n] = acc[r];
    }
  }

  __syncthreads();

  // ---- 3x3/stride2 min/max pool + gt argmax + L1, per pooled pixel ----
  float partial = 0.0f;
  if (tid < PTILE * PTILE) {
    const int pqr = tid >> 3, pqc = tid & 7;
    const int pr = pr0 + pqr, pc = pc0 + pqc;
    float bestGmax = -__builtin_huge_valf(), pAtGmax = 0.0f;
    float bestGmin =  __builtin_huge_valf(), pAtGmin = 0.0f;
    for (int c = 0; c < NF; ++c) {
      float pmx = -__builtin_huge_valf(), pmn = __builtin_huge_valf();
      float gmx = -__builtin_huge_valf(), gmn = __builtin_huge_valf();
#pragma unroll
      for (int dy = 0; dy < 3; ++dy) {
        int cr = 2 * pr - 1 + dy;
        if (cr < 0 || cr >= IMGW) continue;      // -inf border pad
        int lr = 2 * pqr + dy;
#pragma unroll
        for (int dx = 0; dx < 3; ++dx) {
          int cc = 2 * pc - 1 + dx;
          if (cc < 0 || cc >= IMGW) continue;
          int mm = lr * CTILE + (2 * pqc + dx);
          float vp = sConvP[mm * NPAD + c];
          float vg = sConvG[mm * NPAD + c];
          pmx = fmaxf(pmx, vp); pmn = fminf(pmn, vp);
          gmx = fmaxf(gmx, vg); gmn = fminf(gmn, vg);
        }
      }
      if (gmx > bestGmax) { bestGmax = gmx; pAtGmax = pmx; }  // first-index ties
      if (gmn < bestGmin) { bestGmin = gmn; pAtGmin = pmn; }
    }
    partial = fabsf(pAtGmax - bestGmax) + fabsf(pAtGmin - bestGmin);
  }
  sRed[tid] = partial;
  __syncthreads();
  for (int s = BLOCK / 2; s > 0; s >>= 1) {
    if (tid < s) sRed[tid] += sRed[tid + s];
    __syncthreads();
  }
  if (tid == 0) partials[b] = sRed[0];
}

__global__ __launch_bounds__(256) void ctl_reduce(const float* __restrict__ partials,
                                                  float* __restrict__ out) {
  __shared__ float s[256];
  float acc = 0.0f;
  for (int i = threadIdx.x; i < NBLOCKS; i += 256) acc += partials[i];
  s[threadIdx.x] = acc;
  __syncthreads();
  for (int k = 128; k > 0; k >>= 1) {
    if ((int)threadIdx.x < k) s[threadIdx.x] += s[threadIdx.x + k];
    __syncthreads();
  }
  if (threadIdx.x == 0)
    out[0] = s[0] * (1.0f / (16.0f * 256.0f * 256.0f));  // mean_max + mean_min
}

extern "C" void kernel_launch(void* const* d_in, const int* in_sizes, int n_in,
                              void* d_out, int out_size, void* d_ws, size_t ws_size,
                              hipStream_t stream) {
  const float* pred = (const float*)d_in[0];   // [16,1,512,512] f32
  const float* gt   = (const float*)d_in[1];   // [16,1,512,512] f32
  const float* filt = (const float*)d_in[2];   // [14,1,7,7] f32
  float* partials = (float*)d_ws;              // 16384 floats
  ctl_main<<<NBLOCKS, BLOCK, 0, stream>>>(pred, gt, filt, partials);
  ctl_reduce<<<1, 256, 0, stream>>>(partials, (float*)d_out);
}